// ProtGTN_19799799235025
// MI455X (gfx1250) — compile-verified
//
#include <hip/hip_runtime.h>
#include <math.h>

#define N_NODES  100000
#define N_EDGES  3200000
#define G_GRAPHS 256
#define IN_DIM   128
#define HID      6
#define H1       4
#define D1       24          // H1*HID
#define NEG_INF_ENC 0x007FFFFFu
#define INV_SQRT6 0.40824829046386301637f

typedef __attribute__((ext_vector_type(2))) float v2f;
typedef __attribute__((ext_vector_type(8))) float v8f;

static __device__ __forceinline__ unsigned fenc(float f) {
    unsigned u = __float_as_uint(f);
    return (u & 0x80000000u) ? ~u : (u | 0x80000000u);
}
static __device__ __forceinline__ float fdec(unsigned u) {
    return __uint_as_float((u & 0x80000000u) ? (u & 0x7FFFFFFFu) : ~u);
}

// ---------------------------------------------------------------------------
// Workspace init (ws is poisoned once; must re-zero accumulators every call)
// ---------------------------------------------------------------------------
__global__ void init_ws_kernel(float* __restrict__ acc1, float* __restrict__ den1,
                               unsigned* __restrict__ amax1,
                               float* __restrict__ acc2, float* __restrict__ den2,
                               unsigned* __restrict__ amax2,
                               float* __restrict__ psum, unsigned* __restrict__ pmax,
                               float* __restrict__ cnt) {
    int i = blockIdx.x * blockDim.x + threadIdx.x;
    if (i < N_NODES * D1)  acc1[i] = 0.f;
    if (i < N_NODES * H1)  { den1[i] = 0.f; amax1[i] = NEG_INF_ENC; }
    if (i < N_NODES * HID) acc2[i] = 0.f;
    if (i < N_NODES)       { den2[i] = 0.f; amax2[i] = NEG_INF_ENC; }
    if (i < G_GRAPHS * HID){ psum[i] = 0.f; pmax[i] = NEG_INF_ENC; }
    if (i < G_GRAPHS)      cnt[i] = 0.f;
}

// ---------------------------------------------------------------------------
// Layer-1 projections: [N x 128] x [128 x 96] (q1|k1|v1|s1 concat), WMMA f32.
// 8 waves/block, one 16-row tile per wave. Weights staged in LDS,
// pair-interleaved so each lane's B fragment is one ds_load_b64;
// STRIDE % 64 == 32 makes the two half-wave pair-rows hit disjoint banks.
// ---------------------------------------------------------------------------
#define P1_PAIRS  64                       // K/2
#define P1_COLS   96
#define P1_STRIDE (2 * P1_COLS + 32)       // 224 floats; 224 % 64 == 32

__global__ __launch_bounds__(256) void proj1_kernel(
    const float* __restrict__ x,
    const float* __restrict__ qw, const float* __restrict__ qb,
    const float* __restrict__ kw, const float* __restrict__ kb,
    const float* __restrict__ vw, const float* __restrict__ vb,
    const float* __restrict__ sw, const float* __restrict__ sb,
    float* __restrict__ Q, float* __restrict__ K,
    float* __restrict__ V, float* __restrict__ S) {
    __shared__ float Wl[P1_PAIRS * P1_STRIDE];   // 57344 B

    const int tid = threadIdx.x;
    const float* Wm[4] = {qw, kw, vw, sw};
    for (int i = tid; i < P1_PAIRS * P1_COLS; i += 256) {
        const int p = i / P1_COLS, c = i % P1_COLS;
        const int mat = c / D1, cm = c % D1;
        const float* W = Wm[mat];
        float2 w2;
        w2.x = W[(size_t)(2 * p) * D1 + cm];
        w2.y = W[(size_t)(2 * p + 1) * D1 + cm];
        *(float2*)&Wl[p * P1_STRIDE + 2 * c] = w2;
    }
    __syncthreads();

    const int wave = tid >> 5;
    const int lane = tid & 31;
    const int m    = lane & 15;
    const int half = lane >> 4;
    const int tile = blockIdx.x * 8 + wave;
    const bool tvalid = tile < (N_NODES / 16);   // 6250 full tiles
    const int row0 = tvalid ? tile * 16 : 0;

    // A tile: per-lane float2 at K = kk*4 + 2*half (16x4 f32 layout)
    v2f areg[32];
    const float* xr = x + (size_t)(row0 + m) * IN_DIM + 2 * half;
    #pragma unroll
    for (int kk = 0; kk < 32; ++kk) {
        float2 t = *(const float2*)(xr + kk * 4);
        areg[kk].x = t.x; areg[kk].y = t.y;
    }

    const float* Bm[4] = {qb, kb, vb, sb};
    float*       Om[4] = {Q,  K,  V,  S};

    #pragma unroll 1
    for (int t = 0; t < 6; ++t) {
        const int c = t * 16 + m;                // column in concat space [0,96)
        v8f acc = {0.f,0.f,0.f,0.f,0.f,0.f,0.f,0.f};
        #pragma unroll
        for (int kk = 0; kk < 32; ++kk) {
            const int p = kk * 2 + half;         // pair row (k0/2), k0 = kk*4+2*half
            const float2 bw = *(const float2*)&Wl[p * P1_STRIDE + 2 * c];
            v2f b; b.x = bw.x; b.y = bw.y;
            acc = __builtin_amdgcn_wmma_f32_16x16x4_f32(
                false, areg[kk], false, b, (short)0, acc, false, false);
        }
        if (tvalid) {
            const int mat = c / D1, cm = c % D1;
            const float bias = Bm[mat][cm];
            float* O = Om[mat];
            const int rbase = row0 + 8 * half;
            #pragma unroll
            for (int r = 0; r < 8; ++r)
                O[(size_t)(rbase + r) * D1 + cm] = acc[r] + bias;
        }
    }
}

// ---------------------------------------------------------------------------
// Layer-2 projections: [N x 24] x [24 x 24] (q2|k2|v2|s2 concat), WMMA f32.
// Same LDS pair-interleave scheme; STRIDE = 96 floats (96 % 64 == 32).
// ---------------------------------------------------------------------------
#define P2_PAIRS  12                        // K/2 = 24/2
#define P2_COLS   24
#define P2_STRIDE 96

__global__ __launch_bounds__(256) void proj2_kernel(
    const float* __restrict__ h,
    const float* __restrict__ qw, const float* __restrict__ qb,
    const float* __restrict__ kw, const float* __restrict__ kb,
    const float* __restrict__ vw, const float* __restrict__ vb,
    const float* __restrict__ sw, const float* __restrict__ sb,
    float* __restrict__ Q, float* __restrict__ K,
    float* __restrict__ V, float* __restrict__ S) {
    __shared__ float Wl[P2_PAIRS * P2_STRIDE];   // 4608 B

    const int tid = threadIdx.x;
    const float* Wm[4] = {qw, kw, vw, sw};
    for (int i = tid; i < P2_PAIRS * P2_COLS; i += 256) {
        const int p = i / P2_COLS, c = i % P2_COLS;
        const int mat = c / HID, cm = c % HID;
        const float* W = Wm[mat];
        float2 w2;
        w2.x = W[(size_t)(2 * p) * HID + cm];
        w2.y = W[(size_t)(2 * p + 1) * HID + cm];
        *(float2*)&Wl[p * P2_STRIDE + 2 * c] = w2;
    }
    __syncthreads();

    const int wave = tid >> 5;
    const int lane = tid & 31;
    const int m    = lane & 15;
    const int half = lane >> 4;
    const int tile = blockIdx.x * 8 + wave;
    const bool tvalid = tile < (N_NODES / 16);
    const int row0 = tvalid ? tile * 16 : 0;

    v2f areg[6];
    const float* hr = h + (size_t)(row0 + m) * D1 + 2 * half;
    #pragma unroll
    for (int kk = 0; kk < 6; ++kk) {
        float2 t = *(const float2*)(hr + kk * 4);
        areg[kk].x = t.x; areg[kk].y = t.y;
    }

    const float* Bm[4] = {qb, kb, vb, sb};
    float*       Om[4] = {Q,  K,  V,  S};

    #pragma unroll 1
    for (int t = 0; t < 2; ++t) {
        const int c = t * 16 + m;                // concat column in [0,32)
        const int cl = (c < P2_COLS) ? c : (P2_COLS - 1);
        v8f acc = {0.f,0.f,0.f,0.f,0.f,0.f,0.f,0.f};
        #pragma unroll
        for (int kk = 0; kk < 6; ++kk) {
            const int p = kk * 2 + half;
            const float2 bw = *(const float2*)&Wl[p * P2_STRIDE + 2 * cl];
            v2f b; b.x = bw.x; b.y = bw.y;
            acc = __builtin_amdgcn_wmma_f32_16x16x4_f32(
                false, areg[kk], false, b, (short)0, acc, false, false);
        }
        if (tvalid && c < P2_COLS) {
            const int mat = c / HID, cm = c % HID;
            const float bias = Bm[mat][cm];
            float* O = Om[mat];
            const int rbase = row0 + 8 * half;
            #pragma unroll
            for (int r = 0; r < 8; ++r)
                O[(size_t)(rbase + r) * HID + cm] = acc[r] + bias;
        }
    }
}

// ---------------------------------------------------------------------------
// Layer-1 edge pass 1: alpha -> per-dst per-head segment max (ordered-uint)
// ---------------------------------------------------------------------------
__global__ void edge1_max_kernel(const int* __restrict__ ei, const float* __restrict__ ea,
                                 const float* __restrict__ Q, const float* __restrict__ K,
                                 const float* __restrict__ e1w, unsigned* __restrict__ amax) {
    const int e = blockIdx.x * blockDim.x + threadIdx.x;
    if (e >= N_EDGES) return;
    const int s = ei[e];
    const int d = ei[N_EDGES + e];
    const float eav = ea[e];
    const float4* qp = (const float4*)(Q + (size_t)d * D1);
    const float4* kp = (const float4*)(K + (size_t)s * D1);
    const float4* ep = (const float4*)e1w;
    float al[H1] = {0.f, 0.f, 0.f, 0.f};
    #pragma unroll
    for (int i = 0; i < 6; ++i) {
        float4 q = qp[i], k = kp[i], w = ep[i];
        al[(i*4+0)/6] += q.x * (k.x + eav * w.x);
        al[(i*4+1)/6] += q.y * (k.y + eav * w.y);
        al[(i*4+2)/6] += q.z * (k.z + eav * w.z);
        al[(i*4+3)/6] += q.w * (k.w + eav * w.w);
    }
    #pragma unroll
    for (int h = 0; h < H1; ++h)
        atomicMax(&amax[(size_t)d * H1 + h], fenc(al[h] * INV_SQRT6));
}

// ---------------------------------------------------------------------------
// Layer-1 edge pass 2: w = exp(alpha - max); denom += w; acc += w*(v[src]+e)
// ---------------------------------------------------------------------------
__global__ void edge1_acc_kernel(const int* __restrict__ ei, const float* __restrict__ ea,
                                 const float* __restrict__ Q, const float* __restrict__ K,
                                 const float* __restrict__ V, const float* __restrict__ e1w,
                                 const unsigned* __restrict__ amax,
                                 float* __restrict__ den, float* __restrict__ acc) {
    const int e = blockIdx.x * blockDim.x + threadIdx.x;
    if (e >= N_EDGES) return;
    const int s = ei[e];
    const int d = ei[N_EDGES + e];
    const float eav = ea[e];
    const float4* qp = (const float4*)(Q + (size_t)d * D1);
    const float4* kp = (const float4*)(K + (size_t)s * D1);
    const float4* ep = (const float4*)e1w;
    float al[H1] = {0.f, 0.f, 0.f, 0.f};
    #pragma unroll
    for (int i = 0; i < 6; ++i) {
        float4 q = qp[i], k = kp[i], w = ep[i];
        al[(i*4+0)/6] += q.x * (k.x + eav * w.x);
        al[(i*4+1)/6] += q.y * (k.y + eav * w.y);
        al[(i*4+2)/6] += q.z * (k.z + eav * w.z);
        al[(i*4+3)/6] += q.w * (k.w + eav * w.w);
    }
    float wexp[H1];
    #pragma unroll
    for (int h = 0; h < H1; ++h) {
        wexp[h] = __expf(al[h] * INV_SQRT6 - fdec(amax[(size_t)d * H1 + h]));
        atomicAdd(&den[(size_t)d * H1 + h], wexp[h]);
    }
    const float4* vp = (const float4*)(V + (size_t)s * D1);
    float* ad = acc + (size_t)d * D1;
    #pragma unroll
    for (int i = 0; i < 6; ++i) {
        float4 v = vp[i], w = ep[i];
        atomicAdd(&ad[i*4+0], wexp[(i*4+0)/6] * (v.x + eav * w.x));
        atomicAdd(&ad[i*4+1], wexp[(i*4+1)/6] * (v.y + eav * w.y));
        atomicAdd(&ad[i*4+2], wexp[(i*4+2)/6] * (v.z + eav * w.z));
        atomicAdd(&ad[i*4+3], wexp[(i*4+3)/6] * (v.w + eav * w.w));
    }
}

// ---------------------------------------------------------------------------
// Layer-1 finalize: normalize + skip + ELU + LayerNorm(24)
// ---------------------------------------------------------------------------
__global__ void node1_kernel(const float* __restrict__ acc, const float* __restrict__ den,
                             const float* __restrict__ S,
                             const float* __restrict__ g, const float* __restrict__ b,
                             float* __restrict__ hout) {
    const int n = blockIdx.x * blockDim.x + threadIdx.x;
    if (n >= N_NODES) return;
    float o[D1];
    float mu = 0.f;
    #pragma unroll
    for (int j = 0; j < D1; ++j) {
        const float dn = den[(size_t)n * H1 + j / HID] + 1e-16f;
        float v = acc[(size_t)n * D1 + j] / dn + S[(size_t)n * D1 + j];
        v = (v > 0.f) ? v : (__expf(v) - 1.f);   // ELU
        o[j] = v;
        mu += v;
    }
    mu *= (1.f / D1);
    float var = 0.f;
    #pragma unroll
    for (int j = 0; j < D1; ++j) { const float t = o[j] - mu; var += t * t; }
    var *= (1.f / D1);
    const float rs = rsqrtf(var + 1e-5f);
    #pragma unroll
    for (int j = 0; j < D1; ++j)
        hout[(size_t)n * D1 + j] = (o[j] - mu) * rs * g[j] + b[j];
}

// ---------------------------------------------------------------------------
// Layer-2 edge passes (single head, D=6)
// ---------------------------------------------------------------------------
__global__ void edge2_max_kernel(const int* __restrict__ ei, const float* __restrict__ ea,
                                 const float* __restrict__ Q, const float* __restrict__ K,
                                 const float* __restrict__ e2w, unsigned* __restrict__ amax) {
    const int e = blockIdx.x * blockDim.x + threadIdx.x;
    if (e >= N_EDGES) return;
    const int s = ei[e];
    const int d = ei[N_EDGES + e];
    const float eav = ea[e];
    const float* q = Q + (size_t)d * HID;
    const float* k = K + (size_t)s * HID;
    float al = 0.f;
    #pragma unroll
    for (int j = 0; j < HID; ++j) al += q[j] * (k[j] + eav * e2w[j]);
    atomicMax(&amax[d], fenc(al * INV_SQRT6));
}

__global__ void edge2_acc_kernel(const int* __restrict__ ei, const float* __restrict__ ea,
                                 const float* __restrict__ Q, const float* __restrict__ K,
                                 const float* __restrict__ V, const float* __restrict__ e2w,
                                 const unsigned* __restrict__ amax,
                                 float* __restrict__ den, float* __restrict__ acc) {
    const int e = blockIdx.x * blockDim.x + threadIdx.x;
    if (e >= N_EDGES) return;
    const int s = ei[e];
    const int d = ei[N_EDGES + e];
    const float eav = ea[e];
    const float* q = Q + (size_t)d * HID;
    const float* k = K + (size_t)s * HID;
    float al = 0.f;
    #pragma unroll
    for (int j = 0; j < HID; ++j) al += q[j] * (k[j] + eav * e2w[j]);
    const float w = __expf(al * INV_SQRT6 - fdec(amax[d]));
    atomicAdd(&den[d], w);
    const float* v = V + (size_t)s * HID;
    float* ad = acc + (size_t)d * HID;
    #pragma unroll
    for (int j = 0; j < HID; ++j)
        atomicAdd(&ad[j], w * (v[j] + eav * e2w[j]));
}

// ---------------------------------------------------------------------------
// Layer-2 finalize: normalize + skip + ELU + LayerNorm(6)
// ---------------------------------------------------------------------------
__global__ void node2_kernel(const float* __restrict__ acc, const float* __restrict__ den,
                             const float* __restrict__ S,
                             const float* __restrict__ g, const float* __restrict__ b,
                             float* __restrict__ hout) {
    const int n = blockIdx.x * blockDim.x + threadIdx.x;
    if (n >= N_NODES) return;
    const float dn = den[n] + 1e-16f;
    float o[HID];
    float mu = 0.f;
    #pragma unroll
    for (int j = 0; j < HID; ++j) {
        float v = acc[(size_t)n * HID + j] / dn + S[(size_t)n * HID + j];
        v = (v > 0.f) ? v : (__expf(v) - 1.f);
        o[j] = v;
        mu += v;
    }
    mu *= (1.f / HID);
    float var = 0.f;
    #pragma unroll
    for (int j = 0; j < HID; ++j) { const float t = o[j] - mu; var += t * t; }
    var *= (1.f / HID);
    const float rs = rsqrtf(var + 1e-5f);
    #pragma unroll
    for (int j = 0; j < HID; ++j)
        hout[(size_t)n * HID + j] = (o[j] - mu) * rs * g[j] + b[j];
}

// ---------------------------------------------------------------------------
// Segment pooling (mean + max per graph) via atomics
// ---------------------------------------------------------------------------
__global__ void pool_kernel(const float* __restrict__ h2, const int* __restrict__ batch,
                            float* __restrict__ psum, unsigned* __restrict__ pmax,
                            float* __restrict__ cnt) {
    const int n = blockIdx.x * blockDim.x + threadIdx.x;
    if (n >= N_NODES) return;
    const int g = batch[n];
    atomicAdd(&cnt[g], 1.0f);
    #pragma unroll
    for (int j = 0; j < HID; ++j) {
        const float v = h2[(size_t)n * HID + j];
        atomicAdd(&psum[(size_t)g * HID + j], v);
        atomicMax(&pmax[(size_t)g * HID + j], fenc(v));
    }
}

// ---------------------------------------------------------------------------
// Final MLP head: [G,12] -> relu([12,6]) -> [6,1]
// ---------------------------------------------------------------------------
__global__ void final_kernel(const float* __restrict__ psum, const unsigned* __restrict__ pmax,
                             const float* __restrict__ cnt,
                             const float* __restrict__ m1w, const float* __restrict__ m1b,
                             const float* __restrict__ m2w, const float* __restrict__ m2b,
                             float* __restrict__ out) {
    const int g = blockIdx.x * blockDim.x + threadIdx.x;
    if (g >= G_GRAPHS) return;
    float pooled[2 * HID];
    const float c = cnt[g];
    #pragma unroll
    for (int j = 0; j < HID; ++j) pooled[j] = psum[(size_t)g * HID + j] / c;
    #pragma unroll
    for (int j = 0; j < HID; ++j) pooled[HID + j] = fdec(pmax[(size_t)g * HID + j]);
    float o = m2b[0];
    #pragma unroll
    for (int j = 0; j < HID; ++j) {
        float z = m1b[j];
        #pragma unroll
        for (int i = 0; i < 2 * HID; ++i) z += pooled[i] * m1w[i * HID + j];
        z = fmaxf(z, 0.f);
        o += z * m2w[j];
    }
    out[g] = o;
}

// ---------------------------------------------------------------------------
extern "C" void kernel_launch(void* const* d_in, const int* in_sizes, int n_in,
                              void* d_out, int out_size, void* d_ws, size_t ws_size,
                              hipStream_t stream) {
    const float* x     = (const float*)d_in[0];
    const int*   ei    = (const int*)d_in[1];
    const float* ea    = (const float*)d_in[2];
    const int*   batch = (const int*)d_in[3];
    const float *q1w = (const float*)d_in[4],  *q1b = (const float*)d_in[5];
    const float *k1w = (const float*)d_in[6],  *k1b = (const float*)d_in[7];
    const float *v1w = (const float*)d_in[8],  *v1b = (const float*)d_in[9];
    const float *e1w = (const float*)d_in[10];
    const float *s1w = (const float*)d_in[11], *s1b = (const float*)d_in[12];
    const float *q2w = (const float*)d_in[13], *q2b = (const float*)d_in[14];
    const float *k2w = (const float*)d_in[15], *k2b = (const float*)d_in[16];
    const float *v2w = (const float*)d_in[17], *v2b = (const float*)d_in[18];
    const float *e2w = (const float*)d_in[19];
    const float *s2w = (const float*)d_in[20], *s2b = (const float*)d_in[21];
    const float *ln1g = (const float*)d_in[22], *ln1b = (const float*)d_in[23];
    const float *ln2g = (const float*)d_in[24], *ln2b = (const float*)d_in[25];
    const float *m1w = (const float*)d_in[26], *m1b = (const float*)d_in[27];
    const float *m2w = (const float*)d_in[28], *m2b = (const float*)d_in[29];
    float* out = (float*)d_out;

    // workspace carve-out
    float* ws = (float*)d_ws;
    size_t off = 0;
    float* Q1 = ws + off;   off += (size_t)N_NODES * D1;
    float* K1 = ws + off;   off += (size_t)N_NODES * D1;
    float* V1 = ws + off;   off += (size_t)N_NODES * D1;
    float* S1 = ws + off;   off += (size_t)N_NODES * D1;
    float* ACC1 = ws + off; off += (size_t)N_NODES * D1;
    float* DEN1 = ws + off; off += (size_t)N_NODES * H1;
    unsigned* AMAX1 = (unsigned*)(ws + off); off += (size_t)N_NODES * H1;
    float* H1B = ws + off;  off += (size_t)N_NODES * D1;
    float* Q2 = ws + off;   off += (size_t)N_NODES * HID;
    float* K2 = ws + off;   off += (size_t)N_NODES * HID;
    float* V2 = ws + off;   off += (size_t)N_NODES * HID;
    float* S2 = ws + off;   off += (size_t)N_NODES * HID;
    float* ACC2 = ws + off; off += (size_t)N_NODES * HID;
    float* DEN2 = ws + off; off += (size_t)N_NODES;
    unsigned* AMAX2 = (unsigned*)(ws + off); off += (size_t)N_NODES;
    float* H2B = ws + off;  off += (size_t)N_NODES * HID;
    float* PSUM = ws + off; off += (size_t)G_GRAPHS * HID;
    unsigned* PMAX = (unsigned*)(ws + off); off += (size_t)G_GRAPHS * HID;
    float* CNT = ws + off;  off += (size_t)G_GRAPHS;

    const int ib = (N_NODES * D1 + 255) / 256;
    init_ws_kernel<<<ib, 256, 0, stream>>>(ACC1, DEN1, AMAX1, ACC2, DEN2, AMAX2,
                                           PSUM, PMAX, CNT);

    const int ntiles = N_NODES / 16;                 // 6250, exact
    const int pblocks = (ntiles + 7) / 8;            // 8 waves/block
    proj1_kernel<<<pblocks, 256, 0, stream>>>(x, q1w, q1b, k1w, k1b, v1w, v1b,
                                              s1w, s1b, Q1, K1, V1, S1);

    const int eb = (N_EDGES + 255) / 256;
    edge1_max_kernel<<<eb, 256, 0, stream>>>(ei, ea, Q1, K1, e1w, AMAX1);
    edge1_acc_kernel<<<eb, 256, 0, stream>>>(ei, ea, Q1, K1, V1, e1w, AMAX1, DEN1, ACC1);

    const int nb = (N_NODES + 255) / 256;
    node1_kernel<<<nb, 256, 0, stream>>>(ACC1, DEN1, S1, ln1g, ln1b, H1B);

    proj2_kernel<<<pblocks, 256, 0, stream>>>(H1B, q2w, q2b, k2w, k2b, v2w, v2b,
                                              s2w, s2b, Q2, K2, V2, S2);

    edge2_max_kernel<<<eb, 256, 0, stream>>>(ei, ea, Q2, K2, e2w, AMAX2);
    edge2_acc_kernel<<<eb, 256, 0, stream>>>(ei, ea, Q2, K2, V2, e2w, AMAX2, DEN2, ACC2);

    node2_kernel<<<nb, 256, 0, stream>>>(ACC2, DEN2, S2, ln2g, ln2b, H2B);

    pool_kernel<<<nb, 256, 0, stream>>>(H2B, batch, PSUM, PMAX, CNT);

    final_kernel<<<1, 256, 0, stream>>>(PSUM, PMAX, CNT, m1w, m1b, m2w, m2b, out);

    (void)in_sizes; (void)n_in; (void)out_size; (void)ws_size;
}